// GCNII_33603824124480
// MI455X (gfx1250) — compile-verified
//
#include <hip/hip_runtime.h>
#include <hip/hip_bf16.h>
#include <cmath>

#define HDIM 128
#define COUT 64

typedef __attribute__((ext_vector_type(2))) float v2f;
typedef __attribute__((ext_vector_type(8))) float v8f;

// ---------------------------------------------------------------------------
// Setup kernels: degree, dinv, edge norms
// ---------------------------------------------------------------------------
__global__ void init_deg_kernel(float* deg, int n) {
  int i = blockIdx.x * blockDim.x + threadIdx.x;
  if (i < n) deg[i] = 1.0f;  // self-loop weight 1
}

__global__ void accum_deg_kernel(const int* __restrict__ rows,
                                 const float* __restrict__ w,
                                 float* __restrict__ deg, int E) {
  int e = blockIdx.x * blockDim.x + threadIdx.x;
  if (e < E)
    __hip_atomic_fetch_add(&deg[rows[e]], w[e], __ATOMIC_RELAXED,
                           __HIP_MEMORY_SCOPE_AGENT);
}

__global__ void dinv_kernel(float* deg, int n) {  // in-place deg -> dinv
  int i = blockIdx.x * blockDim.x + threadIdx.x;
  if (i < n) {
    float d = deg[i];
    deg[i] = (d > 0.0f) ? (1.0f / sqrtf(d)) : 0.0f;
  }
}

__global__ void norm_kernel(const int* __restrict__ rows,
                            const int* __restrict__ cols,
                            const float* __restrict__ w,
                            const float* __restrict__ dinv,
                            float* __restrict__ norm, int E) {
  int e = blockIdx.x * blockDim.x + threadIdx.x;
  if (e < E) norm[e] = dinv[rows[e]] * w[e] * dinv[cols[e]];
}

// ---------------------------------------------------------------------------
// WMMA GEMM over H=128. Block = 128 threads = 4 waves; each wave owns 32 rows
// (two 16-row A tiles) so every B fragment feeds two v_wmma ops. W staged in
// LDS with a half-wave swizzle (column += 16 for kk%4>=2) so the two
// half-waves of a B-fragment load hit disjoint bank sets (conflict-free).
// MODE_IN : OUT = relu(A@W + bias)
// MODE_SUP: OUT = a_scale*A + b_scale*(A@W)
// MODE_AGG: OUT = a_scale*A + b_scale*(A@W) + dinv[r]^2 * SUP[r][c]
// ---------------------------------------------------------------------------
enum { MODE_IN = 0, MODE_SUP = 1, MODE_AGG = 2 };

template <int MODE, bool GUARD>
__device__ __forceinline__ void gemm_epilogue(
    const v8f (&acc)[2][8], int rowbase, int hi, int m,
    const float* __restrict__ A, float* __restrict__ OUT,
    const float* __restrict__ bias, const float* __restrict__ SUP,
    const float* __restrict__ dinv, float a_scale, float b_scale, int n_rows) {
  #pragma unroll
  for (int mt = 0; mt < 2; ++mt) {
    #pragma unroll
    for (int nt = 0; nt < 8; ++nt) {
      #pragma unroll
      for (int r = 0; r < 8; ++r) {
        const int orow = rowbase + mt * 16 + (hi << 3) + r;
        const int col = nt * 16 + m;
        if (!GUARD || (orow < n_rows)) {
          float v;
          if (MODE == MODE_IN) {
            v = acc[mt][nt][r] + bias[col];
            v = (v > 0.0f) ? v : 0.0f;
          } else {
            const float av = A[(size_t)orow * HDIM + col];
            v = a_scale * av + b_scale * acc[mt][nt][r];
            if (MODE == MODE_AGG) {
              const float dv = dinv[orow];
              v += dv * dv * SUP[(size_t)orow * HDIM + col];
            }
          }
          OUT[(size_t)orow * HDIM + col] = v;
        }
      }
    }
  }
}

template <int MODE>
__global__ __launch_bounds__(128) void gemm128_kernel(
    const float* __restrict__ A, const float* __restrict__ W,
    float* __restrict__ OUT,
    const float* __restrict__ bias,
    const float* __restrict__ SUP,
    const float* __restrict__ dinv,
    float a_scale, float b_scale, int n_rows) {
  __shared__ float sW[HDIM * HDIM];  // 64 KB

  const int tid = threadIdx.x;
  // Stage W with bank swizzle: W[kk][n] -> sW[kk*128 + ((n + ((kk>>1)&1)*16) & 127)]
  for (int j = tid; j < HDIM * HDIM / 4; j += 128) {
    const int idx = j * 4;
    const int kk = idx >> 7;
    const int n = idx & 127;
    const float4 v = *(const float4*)(W + idx);
    const int c = (n + (((kk >> 1) & 1) << 4)) & 127;
    *(float4*)(sW + kk * HDIM + c) = v;
  }
  __syncthreads();

  const int wv = tid >> 5;
  const int lane = tid & 31;
  const int m = lane & 15;       // M (A frag) / N (B frag) index within tile
  const int hi = lane >> 4;      // half-wave: 0 or 1
  const int koff = hi << 1;      // lanes 16..31 carry K+2,K+3
  const int swz = hi << 4;       // B swizzle offset for this half-wave

  const int rowbase = blockIdx.x * 128 + wv * 32;
  const int gr0 = rowbase + m;
  const int gr1 = rowbase + 16 + m;
  const int gl0 = (gr0 < n_rows) ? gr0 : (n_rows - 1);  // keep EXEC all-ones
  const int gl1 = (gr1 < n_rows) ? gr1 : (n_rows - 1);
  const float* arow0 = A + (size_t)gl0 * HDIM;
  const float* arow1 = A + (size_t)gl1 * HDIM;

  v8f acc[2][8];
  #pragma unroll
  for (int mt = 0; mt < 2; ++mt)
    #pragma unroll
    for (int t = 0; t < 8; ++t) acc[mt][t] = v8f{0, 0, 0, 0, 0, 0, 0, 0};

  #pragma unroll 2
  for (int k = 0; k < HDIM; k += 4) {
    const v2f a0 = *(const v2f*)(arow0 + k + koff);
    const v2f a1 = *(const v2f*)(arow1 + k + koff);
    #pragma unroll
    for (int nt = 0; nt < 8; ++nt) {
      const int c = (nt * 16 + m + swz) & 127;
      const float* bp = sW + (k + koff) * HDIM + c;
      v2f b;
      b[0] = bp[0];      // K = k+koff
      b[1] = bp[HDIM];   // K = k+koff+1 (same swizzle)
      acc[0][nt] = __builtin_amdgcn_wmma_f32_16x16x4_f32(
          false, a0, false, b, (short)0, acc[0][nt], false, false);
      acc[1][nt] = __builtin_amdgcn_wmma_f32_16x16x4_f32(
          false, a1, false, b, (short)0, acc[1][nt], false, false);
    }
  }

  // Block-uniform fast path: all 128 rows of this block in range (all blocks
  // except the last) -> branch-free stores, no per-element exec masking.
  if ((int)blockIdx.x * 128 + 128 <= n_rows) {
    gemm_epilogue<MODE, false>(acc, rowbase, hi, m, A, OUT, bias, SUP, dinv,
                               a_scale, b_scale, n_rows);
  } else {
    gemm_epilogue<MODE, true>(acc, rowbase, hi, m, A, OUT, bias, SUP, dinv,
                              a_scale, b_scale, n_rows);
  }
}

// ---------------------------------------------------------------------------
// Edge scatter: one wave per edge; 4 features per lane (float4) = 128 feats.
// agg[col] += norm[e] * support[row]   (native f32 atomics, relaxed/agent)
// ---------------------------------------------------------------------------
__global__ __launch_bounds__(256) void scatter_kernel(
    const int* __restrict__ rows, const int* __restrict__ cols,
    const float* __restrict__ norm, const float* __restrict__ SUP,
    float* __restrict__ AGG, int E) {
  const int e = blockIdx.x * 8 + (threadIdx.x >> 5);
  if (e >= E) return;
  const int lane = threadIdx.x & 31;
  const int r = rows[e];
  const int c = cols[e];
  const float w = norm[e];
  const float4 v = *(const float4*)(SUP + (size_t)r * HDIM + lane * 4);
  float* dst = AGG + (size_t)c * HDIM + lane * 4;
  __hip_atomic_fetch_add(dst + 0, w * v.x, __ATOMIC_RELAXED, __HIP_MEMORY_SCOPE_AGENT);
  __hip_atomic_fetch_add(dst + 1, w * v.y, __ATOMIC_RELAXED, __HIP_MEMORY_SCOPE_AGENT);
  __hip_atomic_fetch_add(dst + 2, w * v.z, __ATOMIC_RELAXED, __HIP_MEMORY_SCOPE_AGENT);
  __hip_atomic_fetch_add(dst + 3, w * v.w, __ATOMIC_RELAXED, __HIP_MEMORY_SCOPE_AGENT);
}

__global__ void relu_kernel(const float* __restrict__ in,
                            float* __restrict__ out, int n4) {
  int i = blockIdx.x * blockDim.x + threadIdx.x;
  if (i < n4) {
    float4 v = ((const float4*)in)[i];
    v.x = fmaxf(v.x, 0.0f);
    v.y = fmaxf(v.y, 0.0f);
    v.z = fmaxf(v.z, 0.0f);
    v.w = fmaxf(v.w, 0.0f);
    ((float4*)out)[i] = v;
  }
}

// ---------------------------------------------------------------------------
// Output: logits = h @ w_out + b_out, then row log-softmax.
// ---------------------------------------------------------------------------
__global__ __launch_bounds__(64) void out_softmax_kernel(
    const float* __restrict__ Hh, const float* __restrict__ Wo,
    const float* __restrict__ bo, float* __restrict__ OUT, int n_rows) {
  __shared__ float sl[COUT];
  const int r = blockIdx.x;
  const int c = threadIdx.x;
  const float* hr = Hh + (size_t)r * HDIM;
  float acc = bo[c];
  #pragma unroll 8
  for (int k = 0; k < HDIM; ++k) acc = fmaf(hr[k], Wo[k * COUT + c], acc);
  sl[c] = acc;
  __syncthreads();
  float mx = -INFINITY;
  #pragma unroll
  for (int j = 0; j < COUT; ++j) mx = fmaxf(mx, sl[j]);
  float s = 0.0f;
  #pragma unroll
  for (int j = 0; j < COUT; ++j) s += expf(sl[j] - mx);
  OUT[(size_t)r * COUT + c] = acc - mx - logf(s);
}

// ---------------------------------------------------------------------------
extern "C" void kernel_launch(void* const* d_in, const int* in_sizes, int n_in,
                              void* d_out, int out_size, void* d_ws,
                              size_t ws_size, hipStream_t stream) {
  const float* x     = (const float*)d_in[0];
  const int*   ei    = (const int*)d_in[1];
  const float* ew    = (const float*)d_in[2];
  const float* w_in  = (const float*)d_in[3];
  const float* b_in  = (const float*)d_in[4];
  const float* ws1   = (const float*)d_in[5];
  const float* ws2   = (const float*)d_in[6];
  const float* w_out = (const float*)d_in[7];
  const float* b_out = (const float*)d_in[8];

  const int N = in_sizes[0] / HDIM;
  const int E = in_sizes[2];
  const int L = in_sizes[5] / (HDIM * HDIM);
  const int* rows = ei;      // edge_index[0]
  const int* cols = ei + E;  // edge_index[1]

  // workspace partition
  char* ws = (char*)d_ws;
  size_t off = 0;
  auto take = [&](size_t bytes) -> float* {
    float* p = (float*)(ws + off);
    off += (bytes + 255) & ~(size_t)255;
    return p;
  };
  const size_t hbytes = (size_t)N * HDIM * sizeof(float);
  float* h    = take(hbytes);
  float* h0   = take(hbytes);
  float* sup  = take(hbytes);
  float* agg  = take(hbytes);
  float* dinv = take((size_t)N * sizeof(float));  // deg, then dinv in-place
  float* norm = take((size_t)E * sizeof(float));
  (void)ws_size; (void)n_in; (void)out_size;

  // ---- GCN normalization (once; reused for all layers) ----
  init_deg_kernel<<<(N + 255) / 256, 256, 0, stream>>>(dinv, N);
  accum_deg_kernel<<<(E + 255) / 256, 256, 0, stream>>>(rows, ew, dinv, E);
  dinv_kernel<<<(N + 255) / 256, 256, 0, stream>>>(dinv, N);
  norm_kernel<<<(E + 255) / 256, 256, 0, stream>>>(rows, cols, ew, dinv, norm, E);

  const int gblocks = (N + 127) / 128;

  // ---- input layer: h = relu(x @ w_in + b_in); h0 = h ----
  gemm128_kernel<MODE_IN><<<gblocks, 128, 0, stream>>>(
      x, w_in, h, b_in, nullptr, nullptr, 0.0f, 1.0f, N);
  hipMemcpyAsync(h0, h, hbytes, hipMemcpyDeviceToDevice, stream);

  // ---- 16 GCNII layers ----
  for (int i = 0; i < L; ++i) {
    const float beta = (float)log(0.5 / (double)(i + 1) + 1.0);
    const float c1 = (1.0f - beta) * (1.0f - 0.1f);  // (1-beta)(1-alpha)
    const float c2 = (1.0f - beta) * 0.1f;           // (1-beta)*alpha
    // support = c1*h + beta*(h @ ws1[i])
    gemm128_kernel<MODE_SUP><<<gblocks, 128, 0, stream>>>(
        h, ws1 + (size_t)i * HDIM * HDIM, sup, nullptr, nullptr, nullptr,
        c1, beta, N);
    // agg = c2*h0 + beta*(h0 @ ws2[i]) + dinv^2 * support   (initial + self-loop)
    gemm128_kernel<MODE_AGG><<<gblocks, 128, 0, stream>>>(
        h0, ws2 + (size_t)i * HDIM * HDIM, agg, nullptr, sup, dinv,
        c2, beta, N);
    // agg[col] += norm * support[row] over all directed edges
    scatter_kernel<<<(E + 7) / 8, 256, 0, stream>>>(rows, cols, norm, sup, agg, E);
    // h = relu(agg)
    relu_kernel<<<((N * HDIM / 4) + 255) / 256, 256, 0, stream>>>(
        agg, h, N * HDIM / 4);
  }

  // ---- output layer + log_softmax ----
  out_softmax_kernel<<<N, COUT, 0, stream>>>(h, w_out, b_out, (float*)d_out, N);
}